// CosineLoss_71494025609906
// MI455X (gfx1250) — compile-verified
//
#include <hip/hip_runtime.h>
#include <hip/hip_bf16.h>
#include <math.h>

#define EPSF 1e-8f

// ---------------- CDNA5 async global->LDS helpers ----------------

typedef int v4i_t __attribute__((ext_vector_type(4)));
typedef __attribute__((address_space(1))) void as1_void_t;
typedef __attribute__((address_space(3))) void as3_void_t;
typedef __attribute__((address_space(1))) v4i_t as1_v4i_t;
typedef __attribute__((address_space(3))) v4i_t as3_v4i_t;
typedef __attribute__((address_space(3))) char as3_char_t;

#if defined(__gfx1250__) && __has_builtin(__builtin_amdgcn_global_load_async_to_lds_b128)
#define ASYNC_VIA_BUILTIN 1
#else
#define ASYNC_VIA_BUILTIN 0
#endif

// Copy 16 bytes (one float4, per active lane) from global memory into LDS via
// the CDNA5 async-DMA path (tracked by ASYNCcnt, no VGPR data movement).
__device__ __forceinline__ void async_copy16_to_lds(const float* gsrc, float* ldst) {
#if ASYNC_VIA_BUILTIN
  __builtin_amdgcn_global_load_async_to_lds_b128(
      (as1_v4i_t*)(as1_void_t*)(void*)const_cast<float*>(gsrc),
      (as3_v4i_t*)(as3_void_t*)ldst,
      /*imm offset*/ 0, /*cpol*/ 0);
#else
  unsigned loff = (unsigned)(unsigned long long)(as3_char_t*)ldst;  // LDS byte offset
  asm volatile("global_load_async_to_lds_b128 %0, %1, off"
               :
               : "v"(loff), "v"(gsrc)
               : "memory");
#endif
}

#if defined(__gfx1250__) && __has_builtin(__builtin_amdgcn_s_wait_asynccnt)
#define WAIT_ASYNCCNT(n)                         \
  do {                                           \
    __builtin_amdgcn_s_wait_asynccnt(n);         \
    asm volatile("" ::: "memory");               \
  } while (0)
#else
#define WAIT_ASYNCCNT(n) asm volatile("s_wait_asynccnt %0" ::"i"(n) : "memory")
#endif

// ---------------- Kernel 1: per-row cosine terms (C == 1000 fast path) ------

__global__ __launch_bounds__(256) void cos_rows_c1000(
    const float* __restrict__ logits, const int* __restrict__ labels,
    float* __restrict__ partials, int N) {
  constexpr int C   = 1000;
  constexpr int C4  = 250;  // float4 per row
  constexpr int K4  = 8;    // ceil(250 / 32) async b128 issues per row per wave
  constexpr int BUF = 1000; // floats per LDS buffer (4000 B, 16B-aligned stride)

  __shared__ __align__(16) float lds[8 * 2 * BUF];  // 8 waves x double buffer
  __shared__ float sred[8];

  const int lane = threadIdx.x & 31;
  const int wv   = threadIdx.x >> 5;  // wave index within block (wave32)
  const long long gw      = (long long)blockIdx.x * 8 + wv;
  const long long gstride = (long long)gridDim.x * 8;

  float* const bufA = &lds[(wv * 2 + 0) * BUF];
  float* const bufB = &lds[(wv * 2 + 1) * BUF];

  float acc = 0.0f;
  long long row = gw;

  // Prime the pipeline: async-stage the first row into bufA.
  if (row < N) {
    const float* g = logits + row * (long long)C;
#pragma unroll
    for (int k = 0; k < K4; ++k) {
      int i4 = k * 32 + lane;
      if (i4 < C4) async_copy16_to_lds(g + i4 * 4, bufA + i4 * 4);
    }
  }

  int cur = 0;
  for (; row < N; row += gstride) {
    float* pb = cur ? bufB : bufA;  // buffer being consumed
    float* nb = cur ? bufA : bufB;  // buffer being prefetched

    const long long nxt = row + gstride;
    if (nxt < N) {
      const float* g = logits + nxt * (long long)C;
#pragma unroll
      for (int k = 0; k < K4; ++k) {
        int i4 = k * 32 + lane;
        if (i4 < C4) async_copy16_to_lds(g + i4 * 4, nb + i4 * 4);
      }
      // 8 new ops outstanding; async ops retire in order, so cnt<=8 means the
      // previous row's 8 transfers have landed in LDS.
      WAIT_ASYNCCNT(K4);
    } else {
      WAIT_ASYNCCNT(0);
    }

    // Sum of squares of this row from LDS (ds_load_b128 per lane).
    float s = 0.0f;
    const float4* p4 = (const float4*)pb;
#pragma unroll
    for (int k = 0; k < K4; ++k) {
      int i4 = k * 32 + lane;
      if (i4 < C4) {
        float4 v = p4[i4];
        s = fmaf(v.x, v.x, s);
        s = fmaf(v.y, v.y, s);
        s = fmaf(v.z, v.z, s);
        s = fmaf(v.w, v.w, s);
      }
    }
    // wave32 butterfly reduction
    s += __shfl_xor(s, 16, 32);
    s += __shfl_xor(s, 8, 32);
    s += __shfl_xor(s, 4, 32);
    s += __shfl_xor(s, 2, 32);
    s += __shfl_xor(s, 1, 32);

    if (lane == 0) {
      int   lab = labels[row];
      float dot = pb[lab];                 // gather from LDS, not global
      float nrm = sqrtf(s);
      acc += 1.0f - dot / fmaxf(nrm, EPSF);
    }
    cur ^= 1;
  }

  if (lane == 0) sred[wv] = acc;
  __syncthreads();
  if (threadIdx.x == 0) {
    float t = 0.0f;
#pragma unroll
    for (int i = 0; i < 8; ++i) t += sred[i];  // fixed order: deterministic
    partials[blockIdx.x] = t;
  }
}

// ---------------- Kernel 1b: generic fallback (any C) ------------------------

__global__ __launch_bounds__(256) void cos_rows_generic(
    const float* __restrict__ logits, const int* __restrict__ labels,
    float* __restrict__ partials, long long N, int C) {
  __shared__ float sred[8];
  const int lane = threadIdx.x & 31;
  const int wv   = threadIdx.x >> 5;
  const long long gw      = (long long)blockIdx.x * 8 + wv;
  const long long gstride = (long long)gridDim.x * 8;

  float acc = 0.0f;
  for (long long row = gw; row < N; row += gstride) {
    const float* g = logits + row * (long long)C;
    float s = 0.0f;
    for (int j = lane; j < C; j += 32) {
      float v = g[j];
      s = fmaf(v, v, s);
    }
    s += __shfl_xor(s, 16, 32);
    s += __shfl_xor(s, 8, 32);
    s += __shfl_xor(s, 4, 32);
    s += __shfl_xor(s, 2, 32);
    s += __shfl_xor(s, 1, 32);
    if (lane == 0) {
      float dot = g[labels[row]];
      acc += 1.0f - dot / fmaxf(sqrtf(s), EPSF);
    }
  }
  if (lane == 0) sred[wv] = acc;
  __syncthreads();
  if (threadIdx.x == 0) {
    float t = 0.0f;
#pragma unroll
    for (int i = 0; i < 8; ++i) t += sred[i];
    partials[blockIdx.x] = t;
  }
}

// ---------------- Kernel 2: deterministic final reduction --------------------

__global__ __launch_bounds__(256) void reduce_partials(
    const float* __restrict__ partials, int np, float invN,
    float* __restrict__ out) {
  __shared__ float sm[256];
  float s = 0.0f;
  for (int i = threadIdx.x; i < np; i += 256) s += partials[i];  // fixed order
  sm[threadIdx.x] = s;
  __syncthreads();
  for (int off = 128; off > 0; off >>= 1) {
    if (threadIdx.x < off) sm[threadIdx.x] += sm[threadIdx.x + off];
    __syncthreads();
  }
  if (threadIdx.x == 0) out[0] = sm[0] * invN;
}

// ---------------- Launch -----------------------------------------------------

extern "C" void kernel_launch(void* const* d_in, const int* in_sizes, int n_in,
                              void* d_out, int out_size, void* d_ws,
                              size_t ws_size, hipStream_t stream) {
  const float* logits = (const float*)d_in[0];
  const int*   labels = (const int*)d_in[1];  // integer input -> const int*
  float*       out    = (float*)d_out;

  const long long NC = in_sizes[0];
  const long long N  = in_sizes[1];
  const int       C  = (int)(NC / N);

  int NB = 2048;  // 2048 blocks x 8 waves = 16384 waves, ~8 rows/wave
  if (ws_size < (size_t)NB * sizeof(float)) {
    NB = (int)(ws_size / sizeof(float));
    if (NB < 1) NB = 1;
    if (NB > 2048) NB = 2048;
  }
  float* partials = (float*)d_ws;

  if (C == 1000) {
    cos_rows_c1000<<<NB, 256, 0, stream>>>(logits, labels, partials, (int)N);
  } else {
    cos_rows_generic<<<NB, 256, 0, stream>>>(logits, labels, partials, N, C);
  }
  reduce_partials<<<1, 256, 0, stream>>>(partials, NB, 1.0f / (float)N, out);
}